// PastEmbedding_67628555043082
// MI455X (gfx1250) — compile-verified
//
#include <hip/hip_runtime.h>
#include <math.h>

typedef __attribute__((ext_vector_type(2))) float v2f;
typedef __attribute__((ext_vector_type(8))) float v8f;

#define D_MODEL 64
#define L_IN    4096
#define L1      2048
#define L2C     1024
#define L3      512

// -ln(10000)/64
#define NEG_LN1E4_OVER_D (-0.14391156816235974f)

// One 16(M) x 16(N) x 256(K) f32 GEMM tile via V_WMMA_F32_16X16X4_F32.
// A comes from an LDS tile laid out as rows of 64 floats, where
// A[row][k] = src[(2*row + (k>>6)) * 64 + (k & 63)]   (conv k=4, stride 2 patches)
// B comes from global conv weights w[(n*64 + c)*4 + j] with k = j*64 + c.
// Per-lane WMMA fragment mapping (CDNA5 ISA 7.12.2):
//   A: M = lane%16, K = kstep + 2*(lane/16) + v      (v = VGPR index 0..1)
//   B: N = lane%16, K = kstep + 2*(lane/16) + v
//   C/D: N = lane%16, M = v + 8*(lane/16)            (v = 0..7)
__device__ __forceinline__ v8f gemm_tile(const float* __restrict__ srcLds, int plmax,
                                         const float* __restrict__ w,
                                         int mt, int n0, int idx, int half) {
  v8f acc = {};
  const int n = n0 + idx;
  const int row = (mt << 4) + idx;
#pragma unroll 4
  for (int k = 0; k < 256; k += 4) {
    int k0 = k + (half << 1);
    int j = k0 >> 6;
    int c = k0 & 63;
    int pl = (row << 1) + j;
    pl = (pl > plmax) ? plmax : pl;   // clamp for garbage M-tile rows (never stored)
    const float* s = srcLds + pl * 64 + c;
    v2f a;
    a[0] = s[0];
    a[1] = s[1];
    const float* wp = w + (((n << 6) + c) << 2) + j;
    v2f bb;
    bb[0] = wp[0];
    bb[1] = wp[4];                    // (n*64 + c + 1)*4 + j
    acc = __builtin_amdgcn_wmma_f32_16x16x4_f32(false, a, false, bb,
                                                (short)0, acc, false, false);
  }
  return acc;
}

__global__ __launch_bounds__(128)
void past_embed_fused(const float* __restrict__ x,
                      const float* __restrict__ tw3, const float* __restrict__ tb3,
                      const float* __restrict__ tw5, const float* __restrict__ tb5,
                      const float* __restrict__ tw7, const float* __restrict__ tb7,
                      const float* __restrict__ tw9, const float* __restrict__ tb9,
                      const float* __restrict__ w1, const float* __restrict__ b1,
                      const float* __restrict__ w2, const float* __restrict__ b2,
                      const float* __restrict__ w3, const float* __restrict__ b3,
                      const float* __restrict__ g1, const float* __restrict__ be1,
                      const float* __restrict__ g2, const float* __restrict__ be2,
                      float* __restrict__ out) {
  // receptive-field chain for 16 final outputs:
  // h0: 142 rows, h1: 70 rows, h2: 34 rows (each row = 64 channels)
  __shared__ float h0s[142 * 64];
  __shared__ float h1s[70 * 64];
  __shared__ float h2s[34 * 64];
  __shared__ float mu[70];
  __shared__ float rsd[70];

  const int b    = blockIdx.y;
  const int t0   = blockIdx.x << 4;    // first of 16 final time-steps
  const int tid  = threadIdx.x;
  const int lane = tid & 31;
  const int wave = tid >> 5;
  const int idx  = lane & 15;
  const int half = lane >> 4;
  const int n0   = wave << 4;          // each wave owns 16 output channels

  // ---------------- stage 0: token embedding + positional embedding ----------------
  const float* xb = x + (size_t)b * L_IN;
  const int p_base = (t0 << 3) - 7;    // h0 global position of local row 0
  for (int i = tid; i < 142 * 64; i += 128) {
    int pl = i >> 6, c = i & 63;
    int p = p_base + pl;
    float v = 0.0f;
    if (p >= 0 && p < L_IN) {
      int ki = c & 3, g = c >> 2;
      int K = 3 + (ki << 1);
      int pad = 1 + ki;
      const float* tw;
      const float* tb;
      if (ki == 0)      { tw = tw3; tb = tb3; }
      else if (ki == 1) { tw = tw5; tb = tb5; }
      else if (ki == 2) { tw = tw7; tb = tb7; }
      else              { tw = tw9; tb = tb9; }
      float acc = tb[g];
      const float* twg = tw + g * K;
      for (int j = 0; j < K; ++j) {
        int q = p + j - pad;
        float xv = (q >= 0 && q < L_IN) ? xb[q] : 0.0f;
        acc += xv * twg[j];
      }
      float freq = expf((float)(c & ~1) * NEG_LN1E4_OVER_D);
      float ang = (float)p * freq;
      acc += (c & 1) ? cosf(ang) : sinf(ang);
      v = acc;
    }
    h0s[i] = v;
  }
  __syncthreads();

  // ---------------- stage 1: conv1 (M=70 pad to 80) -> tanh -> LN ----------------
  for (int mt = 0; mt < 5; ++mt) {
    v8f acc = gemm_tile(h0s, 141, w1, mt, n0, idx, half);
    int n = n0 + idx;
    float bv = b1[n];
    for (int v = 0; v < 8; ++v) {
      int r = (mt << 4) + v + (half << 3);
      if (r < 70) h1s[r * 64 + n] = tanhf(acc[v] + bv);
    }
  }
  __syncthreads();
  if (tid < 70) {
    const float* row = h1s + tid * 64;
    float s = 0.f, s2 = 0.f;
    for (int n = 0; n < 64; ++n) { float v = row[n]; s += v; s2 += v * v; }
    float m = s * (1.0f / 64.0f);
    float var = s2 * (1.0f / 64.0f) - m * m;
    mu[tid] = m;
    rsd[tid] = rsqrtf(var + 1e-5f);
  }
  __syncthreads();
  {
    const int q_base = (t0 << 2) - 3;  // h1 global position of local row 0
    for (int i = tid; i < 70 * 64; i += 128) {
      int m = i >> 6, n = i & 63;
      int q = q_base + m;
      float v = 0.0f;
      if (q >= 0 && q < L1)
        v = (h1s[i] - mu[m]) * rsd[m] * g1[n] + be1[n];
      h1s[i] = v;                      // OOB rows become conv2's zero padding
    }
  }
  __syncthreads();

  // ---------------- stage 2: conv2 (M=34 pad to 48) -> tanh -> LN ----------------
  for (int mt = 0; mt < 3; ++mt) {
    v8f acc = gemm_tile(h1s, 69, w2, mt, n0, idx, half);
    int n = n0 + idx;
    float bv = b2[n];
    for (int v = 0; v < 8; ++v) {
      int r = (mt << 4) + v + (half << 3);
      if (r < 34) h2s[r * 64 + n] = tanhf(acc[v] + bv);
    }
  }
  __syncthreads();
  if (tid < 34) {
    const float* row = h2s + tid * 64;
    float s = 0.f, s2 = 0.f;
    for (int n = 0; n < 64; ++n) { float v = row[n]; s += v; s2 += v * v; }
    float m = s * (1.0f / 64.0f);
    float var = s2 * (1.0f / 64.0f) - m * m;
    mu[tid] = m;
    rsd[tid] = rsqrtf(var + 1e-5f);
  }
  __syncthreads();
  {
    const int r_base = (t0 << 1) - 1;  // h2 global position of local row 0
    for (int i = tid; i < 34 * 64; i += 128) {
      int m = i >> 6, n = i & 63;
      int r = r_base + m;
      float v = 0.0f;
      if (r >= 0 && r < L2C)
        v = (h2s[i] - mu[m]) * rsd[m] * g2[n] + be2[n];
      h2s[i] = v;                      // OOB rows become conv3's zero padding
    }
  }
  __syncthreads();

  // ---------------- stage 3: conv3 (M=16) -> tanh -> global output ----------------
  {
    v8f acc = gemm_tile(h2s, 33, w3, 0, n0, idx, half);
    int n = n0 + idx;
    float bv = b3[n];
    float* ob = out + ((size_t)b * L3 + t0) * 64;
    for (int v = 0; v < 8; ++v) {
      int r = v + (half << 3);
      ob[r * 64 + n] = tanhf(acc[v] + bv);
    }
  }
}

extern "C" void kernel_launch(void* const* d_in, const int* in_sizes, int n_in,
                              void* d_out, int out_size, void* d_ws, size_t ws_size,
                              hipStream_t stream) {
  (void)in_sizes; (void)n_in; (void)d_ws; (void)ws_size; (void)out_size;
  const float* x   = (const float*)d_in[0];
  const float* tw3 = (const float*)d_in[1];
  const float* tb3 = (const float*)d_in[2];
  const float* tw5 = (const float*)d_in[3];
  const float* tb5 = (const float*)d_in[4];
  const float* tw7 = (const float*)d_in[5];
  const float* tb7 = (const float*)d_in[6];
  const float* tw9 = (const float*)d_in[7];
  const float* tb9 = (const float*)d_in[8];
  const float* w1  = (const float*)d_in[9];
  const float* b1  = (const float*)d_in[10];
  const float* w2  = (const float*)d_in[11];
  const float* b2  = (const float*)d_in[12];
  const float* w3  = (const float*)d_in[13];
  const float* b3  = (const float*)d_in[14];
  const float* g1  = (const float*)d_in[15];
  const float* be1 = (const float*)d_in[16];
  const float* g2  = (const float*)d_in[17];
  const float* be2 = (const float*)d_in[18];
  float* out = (float*)d_out;

  dim3 grid(L3 / 16, 128);   // 32 time-tiles x 128 batches = 4096 workgroups
  dim3 block(128);           // 4 wave32: one 16-channel WMMA column tile per wave
  past_embed_fused<<<grid, block, 0, stream>>>(
      x, tw3, tb3, tw5, tb5, tw7, tb7, tw9, tb9,
      w1, b1, w2, b2, w3, b3, g1, be1, g2, be2, out);
}